// HeteroDotProductPredictor_68092411510983
// MI455X (gfx1250) — compile-verified
//
#include <hip/hip_runtime.h>

// CDNA5 / gfx1250: wave32, WMMA (not MFMA).
typedef float v2f __attribute__((ext_vector_type(2)));
typedef float v8f __attribute__((ext_vector_type(8)));

#define FEAT_D        128
#define EDGES_PER_WAVE 16
#define WAVES_PER_BLK   8
#define BLOCK_THREADS (WAVES_PER_BLK * 32)

// For each edge e: score[e] = dot(h_src[src_idx[e]], h_dst[dst_idx[e]])
// One wave handles 16 edges. The 16 dot products are computed as the diagonal
// of C += A(16x4) * B(4x16) accumulated over K = 0..127 in steps of 4 with
// V_WMMA_F32_16X16X4_F32.
//
// f32 A(16x4) VGPR layout (ISA 7.12.2): lane L (L<16) -> M=L holds K=k0,k0+1 ;
// lane L+16 -> M=L holds K=k0+2,k0+3.  B(4x16) is the mirrored layout
// (lane L -> N=L, same K split).  So per K-step each lane does one b64 load
// from "its" src row and one b64 load from "its" dst row at float offset
// k0 + 2*(lane>>4).  Per-row access is contiguous 16B strips -> full
// cacheline use; both tables (102 MB) sit in the 192 MB L2 across replays.
__global__ __launch_bounds__(BLOCK_THREADS)
void edge_udotv_wmma_f32(const float* __restrict__ h_src,
                         const float* __restrict__ h_dst,
                         const int*   __restrict__ src_idx,
                         const int*   __restrict__ dst_idx,
                         float*       __restrict__ out,
                         int E)
{
    __shared__ float lds[WAVES_PER_BLK][32 * 8];

    const int tid  = threadIdx.x;
    const int wave = tid >> 5;
    const int lane = tid & 31;
    const int m    = lane & 15;   // which of the wave's 16 edges this lane feeds
    const int h    = lane >> 4;   // K-half selector (0: K+0,K+1 / 1: K+2,K+3)

    const long long ebase = ((long long)blockIdx.x * WAVES_PER_BLK + wave) * EDGES_PER_WAVE;

    // Clamp tail edges instead of masking: EXEC must stay all-ones for WMMA.
    long long eg = ebase + m;
    if (eg >= E) eg = (long long)E - 1;

    const long long srow = (long long)src_idx[eg] * FEAT_D;
    const long long trow = (long long)dst_idx[eg] * FEAT_D;
    const float* sp = h_src + srow + 2 * h;
    const float* tp = h_dst + trow + 2 * h;

    v8f acc = {0.f, 0.f, 0.f, 0.f, 0.f, 0.f, 0.f, 0.f};

#pragma unroll
    for (int k0 = 0; k0 < FEAT_D; k0 += 4) {
        v2f a = *(const v2f*)(sp + k0);   // global_load_b64 (A: src rows)
        v2f b = *(const v2f*)(tp + k0);   // global_load_b64 (B: dst rows)
        // D = A x B + C ; emits v_wmma_f32_16x16x4_f32
        acc = __builtin_amdgcn_wmma_f32_16x16x4_f32(
            /*neg_a=*/false, a, /*neg_b=*/false, b,
            /*c_mod=*/(short)0, acc, /*reuse_a=*/false, /*reuse_b=*/false);
    }

    // Spill accumulator to LDS, then lanes 0..15 pick the diagonal C[e][e].
    // C/D layout: VGPR v holds (M=v, N=lane) for lanes 0-15 and (M=v+8,
    // N=lane-16) for lanes 16-31.  Diagonal element for edge e lives at:
    //   e < 8 : lane=e,     vgpr=e     -> lds index 9*e
    //   e >= 8: lane=e+16,  vgpr=e-8   -> lds index 9*e + 120
#pragma unroll
    for (int i = 0; i < 8; ++i)
        lds[wave][lane * 8 + i] = acc[i];

    __syncthreads();

    if (lane < 16) {
        const int e   = lane;
        const int idx = (e < 8) ? (9 * e) : (9 * e + 120);
        const long long eo = ebase + e;
        if (eo < E)
            out[eo] = lds[wave][idx];
    }
}

extern "C" void kernel_launch(void* const* d_in, const int* in_sizes, int n_in,
                              void* d_out, int out_size, void* d_ws, size_t ws_size,
                              hipStream_t stream) {
    const float* h_src   = (const float*)d_in[0];
    const float* h_dst   = (const float*)d_in[1];
    const int*   src_idx = (const int*)d_in[2];   // harness contract: integer -> const int*
    const int*   dst_idx = (const int*)d_in[3];
    float*       out     = (float*)d_out;

    const int E = in_sizes[2];                    // 625000 edges
    const int edges_per_block = WAVES_PER_BLK * EDGES_PER_WAVE;  // 128
    const int blocks = (E + edges_per_block - 1) / edges_per_block;

    edge_udotv_wmma_f32<<<dim3(blocks), dim3(BLOCK_THREADS), 0, stream>>>(
        h_src, h_dst, src_idx, dst_idx, out, E);
}